// SimpleGeluEmbed_9792525435301
// MI455X (gfx1250) — compile-verified
//
#include <hip/hip_runtime.h>
#include <hip/hip_bf16.h>
#include <math.h>

// ---- problem constants (match reference) ----
#define TOKENS 20
#define DIM    128
#define CELLS_PER_WAVE  16
#define WAVES_PER_BLOCK 4
#define THREADS_PER_BLOCK (WAVES_PER_BLOCK * 32)
#define CELLS_PER_BLOCK (CELLS_PER_WAVE * WAVES_PER_BLOCK)  // 64
#define LDS_STRIDE 132  // 128 dims padded to 132 floats -> conflict-free ds_load_b64

typedef __attribute__((ext_vector_type(2))) float v2f;
typedef __attribute__((ext_vector_type(8))) float v8f;

__device__ __forceinline__ float gelu_exact(float v) {
    // exact erf GELU: 0.5*x*(1+erf(x/sqrt(2)))
    return 0.5f * v * (1.0f + erff(v * 0.70710678118654752440f));
}

__global__ __launch_bounds__(THREADS_PER_BLOCK)
void SimpleGeluEmbed_kernel(const int* __restrict__ x,
                            const float* __restrict__ emb,
                            const float* __restrict__ W,
                            const float* __restrict__ bias_p,
                            float* __restrict__ out) {
    __shared__ float agg[WAVES_PER_BLOCK * CELLS_PER_WAVE * LDS_STRIDE];

    const int lane = threadIdx.x & 31;
    const int wave = threadIdx.x >> 5;
    const int waveCellBase = blockIdx.x * CELLS_PER_BLOCK + wave * CELLS_PER_WAVE;

    float* aggWave = agg + wave * (CELLS_PER_WAVE * LDS_STRIDE);
    const float4* emb4 = reinterpret_cast<const float4*>(emb);

    // ---------------- Phase 1: gather + mean + GELU -> LDS ----------------
    // Wave handles one cell at a time: 32 lanes read one embedding row as a
    // single coalesced 512B burst (lane owns dims 4*lane..4*lane+3).
    for (int m = 0; m < CELLS_PER_WAVE; ++m) {
        const int cell = waveCellBase + m;
        const int* xp = x + cell * TOKENS;   // uniform per wave -> s_load
        float ax = 0.f, ay = 0.f, az = 0.f, aw = 0.f;
#pragma unroll
        for (int t = 0; t < TOKENS; ++t) {
            const int idx = xp[t];
            const float4 v = emb4[idx * (DIM / 4) + lane];  // global_load_b128
            ax += v.x; ay += v.y; az += v.z; aw += v.w;
        }
        const float inv = 1.0f / (float)TOKENS;
        float4 h;
        h.x = gelu_exact(ax * inv);
        h.y = gelu_exact(ay * inv);
        h.z = gelu_exact(az * inv);
        h.w = gelu_exact(aw * inv);
        *reinterpret_cast<float4*>(aggWave + m * LDS_STRIDE + lane * 4) = h;
    }

    __syncthreads();  // make wave's LDS writes visible across lanes

    // ---------------- Phase 2: 16x128 GEMV via V_WMMA_F32_16X16X4_F32 ------
    // A (16x4 f32) fragment: lanes 0-15 -> M=lane, K={0,1}; lanes 16-31 ->
    // M=lane-16, K={2,3}. B (4x16) replicated across N: every D column = logit.
    const int koff  = (lane & 16) ? 2 : 0;
    const int cellm = lane & 15;
    const float* arow = aggWave + cellm * LDS_STRIDE + koff;  // 8B aligned

    v8f c = {};
#pragma unroll
    for (int j = 0; j < 32; ++j) {
        const v2f a  = *reinterpret_cast<const v2f*>(arow + 4 * j);      // ds_load_b64
        const v2f bw = *reinterpret_cast<const v2f*>(W + 4 * j + koff);  // global_load_b64
        // 8 args: (neg_a, A, neg_b, B, c_mod, C, reuse_a, reuse_b)
        c = __builtin_amdgcn_wmma_f32_16x16x4_f32(false, a, false, bw,
                                                  (short)0, c, false, false);
    }

    // D layout: lane 0 holds D[0..7][0] in c[0..7]; lane 16 holds D[8..15][0].
    const float bb = bias_p[0];
    if (lane == 0 || lane == 16) {
        float* op = out + waveCellBase + ((lane >> 4) << 3);
        float4 lo = { c[0] + bb, c[1] + bb, c[2] + bb, c[3] + bb };
        float4 hi = { c[4] + bb, c[5] + bb, c[6] + bb, c[7] + bb };
        *reinterpret_cast<float4*>(op)     = lo;
        *reinterpret_cast<float4*>(op + 4) = hi;
    }
}

extern "C" void kernel_launch(void* const* d_in, const int* in_sizes, int n_in,
                              void* d_out, int out_size, void* d_ws, size_t ws_size,
                              hipStream_t stream) {
    const int*   x   = (const int*)d_in[0];    // [B,R,C,T] int32
    const float* emb = (const float*)d_in[1];  // [VOCAB, DIM] f32
    const float* W   = (const float*)d_in[2];  // [DIM, 1] f32
    const float* b   = (const float*)d_in[3];  // [1] f32
    float*       out = (float*)d_out;          // [B,R,C] f32

    const int n_cells = out_size;              // 80000, divisible by 64
    const int blocks  = n_cells / CELLS_PER_BLOCK;

    SimpleGeluEmbed_kernel<<<blocks, THREADS_PER_BLOCK, 0, stream>>>(
        x, emb, W, b, out);
}